// LSTM4RNMT_17454747090988
// MI455X (gfx1250) — compile-verified
//
#include <hip/hip_runtime.h>
#include <math.h>
#include <stdint.h>

// Problem constants
#define Bsz 64
#define Tlen 512
#define Idim 512
#define Odim 512
#define N4O 2048       // 4 * O
#define EPSV 1e-5f

#define KSL 256                 // K-slice handled by one GEMM plane
#define LDS_ROWS 32             // M rows staged per block
#define LDS_STRIDE (KSL + 4)    // +4 floats padding -> conflict-free ds_load_b64

typedef __attribute__((ext_vector_type(2))) float v2f;
typedef __attribute__((ext_vector_type(8))) float v8f;

// ---------------------------------------------------------------------------
// CDNA5 async memory->LDS staging (ASYNCcnt path, bypasses VGPR returns)
// ---------------------------------------------------------------------------
__device__ __forceinline__ void async_ld_b128(unsigned lds_addr, const float* gaddr) {
  asm volatile("global_load_async_to_lds_b128 %0, %1, off"
               :: "v"(lds_addr), "v"(gaddr) : "memory");
}
__device__ __forceinline__ void async_wait0() {
  asm volatile("s_wait_asynccnt 0" ::: "memory");
}

// ---------------------------------------------------------------------------
// One GEMM "plane": partial(64 x 2048) = A(64 x 256) @ W[kW0:kW0+256, :]
// 256 waves per plane (32 blocks x 8 waves). Each wave: 32x16 output slab
// (2 M-tiles share each B fragment), 128 x v_wmma_f32_16x16x4_f32.
// A tile (32 rows x 256 cols) staged in LDS via async loads.
// ---------------------------------------------------------------------------
__device__ __forceinline__ void gemm_plane_body(
    const float* __restrict__ Asrc,   // A rows 0..63, cols = this 256-slice
    size_t Astride,                   // A row stride in floats
    const float* __restrict__ W,      // full W (1024 x 2048)
    int kW0,                          // W row base of this slice
    float* __restrict__ outp,         // partial plane (64 x 2048)
    int blk)                          // 0..31 within plane
{
  __shared__ __align__(16) float As[LDS_ROWS * LDS_STRIDE];
  const int tid  = threadIdx.x;
  const int lane = tid & 31;
  const int wave = blk * 8 + (tid >> 5);   // 0..255 within plane
  const int tn   = wave & 127;             // N tile
  const int mh   = wave >> 7;              // M half (uniform per block: 8 | 128)
  const int l16  = lane & 15;
  const int half = lane >> 4;              // K sub-pair select per WMMA layout

  // ---- stage A tile (32 x 256 f32 = 32 KB) into LDS asynchronously ----
  const unsigned asbase = (unsigned)(uintptr_t)(void*)As;
  for (int i = tid; i < LDS_ROWS * (KSL / 4); i += 256) {
    const int row = i >> 6;      // i / (KSL/4)
    const int c4  = i & 63;
    async_ld_b128(asbase + (unsigned)((row * LDS_STRIDE + c4 * 4) * 4),
                  Asrc + (size_t)(mh * LDS_ROWS + row) * Astride + c4 * 4);
  }
  async_wait0();
  __syncthreads();

  const int col = tn * 16 + l16;
  const float* __restrict__ Wp = W + (size_t)kW0 * N4O + col;

  v8f acc0 = {};
  v8f acc1 = {};
  for (int kk = 0; kk < KSL; kk += 4) {
    const int ka = kk + half * 2;
    v2f a0, a1, bf;
    // A fragments from LDS (row-padded: bank-conflict free b64 reads)
    a0[0] = As[l16 * LDS_STRIDE + ka];
    a0[1] = As[l16 * LDS_STRIDE + ka + 1];
    a1[0] = As[(16 + l16) * LDS_STRIDE + ka];
    a1[1] = As[(16 + l16) * LDS_STRIDE + ka + 1];
    // B fragment: W[kW0+ka][col], W[kW0+ka+1][col] (shared by both M tiles)
    bf[0] = Wp[(size_t)ka * N4O];
    bf[1] = Wp[(size_t)(ka + 1) * N4O];
    acc0 = __builtin_amdgcn_wmma_f32_16x16x4_f32(false, a0, false, bf,
                                                 (short)0, acc0, false, false);
    acc1 = __builtin_amdgcn_wmma_f32_16x16x4_f32(false, a1, false, bf,
                                                 (short)0, acc1, false, false);
  }

  // D layout: VGPR r -> M = base + half*8 + r, N = col
  float* o0 = outp + (size_t)(mh * 32 + half * 8) * N4O + col;
  float* o1 = o0 + (size_t)16 * N4O;
#pragma unroll
  for (int r = 0; r < 8; ++r) {
    o0[(size_t)r * N4O] = acc0[r];
    o1[(size_t)r * N4O] = acc1[r];
  }
}

// ---------------------------------------------------------------------------
// x-projection (recurrence-independent): chunk of `cnt` timesteps.
// grid = cnt * 64 blocks: [tloc | s | blk]
// ---------------------------------------------------------------------------
__global__ __launch_bounds__(256) void lstm_gemm_x(const float* __restrict__ x,
                                                   const float* __restrict__ W,
                                                   float* __restrict__ combX,
                                                   int t0) {
  const int bi   = blockIdx.x;
  const int tloc = bi >> 6;
  const int rest = bi & 63;
  const int s    = rest >> 5;
  const int blk  = rest & 31;
  const int t    = t0 + tloc;
  gemm_plane_body(x + (size_t)t * Idim + s * KSL, (size_t)Tlen * Idim,
                  W, s * KSL,
                  combX + ((size_t)tloc * 2 + s) * Bsz * N4O, blk);
}

// ---------------------------------------------------------------------------
// h-projection (serial critical path): K = 512 split into 2 planes.
// grid = 64 blocks: [s | blk]
// ---------------------------------------------------------------------------
__global__ __launch_bounds__(256) void lstm_gemm_h(const float* __restrict__ Z,
                                                   const float* __restrict__ W,
                                                   float* __restrict__ combH) {
  const int s   = blockIdx.x >> 5;
  const int blk = blockIdx.x & 31;
  gemm_plane_body(Z + s * KSL, (size_t)Odim,
                  W, Idim + s * KSL,
                  combH + (size_t)s * Bsz * N4O, blk);
}

// ---------------------------------------------------------------------------
// Init: h = init_hx (broadcast), c = init_cx (broadcast)
// ---------------------------------------------------------------------------
__global__ __launch_bounds__(256) void lstm_init(const float* __restrict__ hx,
                                                 const float* __restrict__ cx,
                                                 float* __restrict__ Z,
                                                 float* __restrict__ c) {
  const int b = blockIdx.x, tid = threadIdx.x;
  for (int i = tid; i < Odim; i += 256) {
    Z[b * Odim + i] = hx[i];
    c[b * Odim + i] = cx[i];
  }
}

// ---------------------------------------------------------------------------
// Sum 4 partial planes + bias, joint LayerNorm over 2048/row, gates, state.
// One block per batch row.
// ---------------------------------------------------------------------------
__global__ __launch_bounds__(256) void lstm_pointwise(
    const float* __restrict__ combH,
    const float* __restrict__ combXslot,   // 2 planes for this timestep
    const float* __restrict__ bias,
    const float* __restrict__ ln_w,
    const float* __restrict__ ln_b,
    float* __restrict__ c,
    float* __restrict__ Z,                 // h buffer (64 x 512)
    float* __restrict__ out,
    int t) {
  __shared__ float ssum[256];
  __shared__ float ssq[256];
  const int b = blockIdx.x;
  const int tid = threadIdx.x;
  const float* p0 = combH + (size_t)b * N4O;
  const float* p1 = p0 + (size_t)Bsz * N4O;
  const float* q0 = combXslot + (size_t)b * N4O;
  const float* q1 = q0 + (size_t)Bsz * N4O;

  float vv[8];
  float s = 0.f, q = 0.f;
#pragma unroll
  for (int j = 0; j < 8; ++j) {
    const int i = tid + j * 256;
    const float v = p0[i] + p1[i] + q0[i] + q1[i] + bias[i];
    vv[j] = v;
    s += v;
    q += v * v;
  }
  ssum[tid] = s;
  ssq[tid] = q;
  __syncthreads();
  for (int off = 128; off > 0; off >>= 1) {
    if (tid < off) {
      ssum[tid] += ssum[tid + off];
      ssq[tid]  += ssq[tid + off];
    }
    __syncthreads();
  }
  const float mu   = ssum[0] * (1.0f / N4O);
  const float var  = ssq[0] * (1.0f / N4O) - mu * mu;
  const float rstd = rsqrtf(var + EPSV);

#pragma unroll
  for (int oo = 0; oo < 2; ++oo) {
    const int o = tid + oo * 256;
    float g[4];
#pragma unroll
    for (int gi = 0; gi < 4; ++gi) {
      g[gi] = (vv[gi * 2 + oo] - mu) * rstd * ln_w[gi * Odim + o] + ln_b[gi * Odim + o];
    }
    const float ig  = 1.f / (1.f + expf(-g[0]));
    const float fg  = 1.f / (1.f + expf(-g[1]));
    const float og  = 1.f / (1.f + expf(-g[2]));
    const float hid = tanhf(g[3]);
    const float cn  = fg * c[b * Odim + o] + ig * hid;
    const float h   = og * cn;            // reference: h = og * c (no tanh)
    c[b * Odim + o] = cn;
    Z[b * Odim + o] = h;                  // feeds next step's gemm_h
    out[((size_t)b * Tlen + t) * Odim + o] = h;
  }
}

// ---------------------------------------------------------------------------
extern "C" void kernel_launch(void* const* d_in, const int* in_sizes, int n_in,
                              void* d_out, int out_size, void* d_ws, size_t ws_size,
                              hipStream_t stream) {
  const float* x    = (const float*)d_in[0];  // (B, T, I)
  const float* W    = (const float*)d_in[1];  // (I+O, 4O)
  const float* bias = (const float*)d_in[2];  // (4O,)
  const float* ln_w = (const float*)d_in[3];  // (4, O)
  const float* ln_b = (const float*)d_in[4];  // (4, O)
  const float* hx   = (const float*)d_in[5];  // (1, O)
  const float* cx   = (const float*)d_in[6];  // (1, O)
  float* out = (float*)d_out;                 // (B, T, O)

  // workspace layout (floats)
  float* Z     = (float*)d_ws;                 // h       : 64 x 512
  float* c     = Z + Bsz * Odim;               // c       : 64 x 512
  float* combH = c + Bsz * Odim;               // 2 planes: 2 x 64 x 2048
  float* combX = combH + 2 * (size_t)Bsz * N4O;

  // chunk size for the recurrence-independent x-projection ring buffer,
  // derived deterministically from ws_size
  const size_t fixedB   = ((size_t)2 * Bsz * Odim + 2 * (size_t)Bsz * N4O) * 4;
  const size_t per_slot = (size_t)2 * Bsz * N4O * 4;   // 1 MB per timestep slot
  int CH = 1;
  if (ws_size > fixedB + per_slot) {
    size_t m = (ws_size - fixedB) / per_slot;
    CH = (m > 64) ? 64 : (int)m;
  }

  lstm_init<<<Bsz, 256, 0, stream>>>(hx, cx, Z, c);
  for (int t = 0; t < Tlen; ++t) {
    if (t % CH == 0) {
      const int cnt = (Tlen - t < CH) ? (Tlen - t) : CH;
      lstm_gemm_x<<<cnt * 64, 256, 0, stream>>>(x, W, combX, t);
    }
    lstm_gemm_h<<<64, 256, 0, stream>>>(Z, W, combH);
    lstm_pointwise<<<Bsz, 256, 0, stream>>>(
        combH, combX + (size_t)(t % CH) * 2 * Bsz * N4O,
        bias, ln_w, ln_b, c, Z, out, t);
  }
}